// QuantizeInterpolatedEmbedding_63866163692089
// MI455X (gfx1250) — compile-verified
//
#include <hip/hip_runtime.h>
#include <hip/hip_bf16.h>
#include <math.h>

// Problem constants (match reference)
#define BB   32
#define CC   512
#define TT   256
#define VV   8192
#define QQ   64

typedef float v2f __attribute__((ext_vector_type(2)));
typedef float v8f __attribute__((ext_vector_type(8)));

#define LDA 516   // padded LDS row stride (516 % 64 = 4 -> conflict-free fragment reads)

// Async global->LDS copy of 16 bytes (gfx1250 GLOBAL_LOAD_ASYNC_TO_LDS_B128).
// ldsOff = byte offset within workgroup LDS (low 32 bits of generic pointer).
__device__ __forceinline__ void async_copy_b128(uint32_t ldsOff, const void* gptr) {
    uint64_t ga = (uint64_t)(uintptr_t)gptr;
    asm volatile("global_load_async_to_lds_b128 %0, %1, off"
                 :: "v"(ldsOff), "v"(ga) : "memory");
}
__device__ __forceinline__ void async_wait_all() {
    asm volatile("s_wait_asynccnt 0x0" ::: "memory");
}

// ---------------------------------------------------------------------------
// Prep: area-downsample z [B,C,T] -> X1 [B*Q, C] row-major (mean of 4)
// ---------------------------------------------------------------------------
__global__ __launch_bounds__(256) void downsample_kernel(
    const float* __restrict__ z, float* __restrict__ x1)
{
    int idx = blockIdx.x * 256 + threadIdx.x;           // over (B*Q)*C
    if (idx >= BB * QQ * CC) return;
    int c    = idx & (CC - 1);
    int nrow = idx >> 9;                                // b*Q + tq
    int b    = nrow >> 6;
    int tq   = nrow & (QQ - 1);
    const float* p = z + ((size_t)b * CC + c) * TT + tq * 4;
    x1[idx] = 0.25f * (p[0] + p[1] + p[2] + p[3]);
}

// ---------------------------------------------------------------------------
// Prep: transpose z [B,C,T] -> X2 [B*T, C] row-major
// ---------------------------------------------------------------------------
__global__ __launch_bounds__(256) void transpose_kernel(
    const float* __restrict__ z, float* __restrict__ x2)
{
    int idx = blockIdx.x * 256 + threadIdx.x;           // over (B*T)*C
    if (idx >= BB * TT * CC) return;
    int c    = idx & (CC - 1);
    int nrow = idx >> 9;                                // b*T + t
    int b    = nrow >> 8;
    int t    = nrow & (TT - 1);
    x2[idx] = z[((size_t)b * CC + c) * TT + t];
}

// ---------------------------------------------------------------------------
// Prep: codebook squared row norms cb_sq[v] = sum_c cb[v][c]^2 (wave per row)
// ---------------------------------------------------------------------------
__global__ __launch_bounds__(256) void cbsq_kernel(
    const float* __restrict__ cb, float* __restrict__ cbsq)
{
    int wave = threadIdx.x >> 5, lane = threadIdx.x & 31;
    int v = blockIdx.x * 8 + wave;
    if (v >= VV) return;
    const float* row = cb + (size_t)v * CC;
    float s = 0.0f;
    for (int c = lane; c < CC; c += 32) { float x = row[c]; s += x * x; }
    #pragma unroll
    for (int off = 16; off >= 1; off >>= 1) s += __shfl_xor(s, off, 32);
    if (lane == 0) cbsq[v] = s;
}

// ---------------------------------------------------------------------------
// Distance + argmin: score[n,v] = cb_sq[v] - 2 * dot(X[n], cb[v])
// (row-constant ||x||^2 / sum(x) terms cannot change the argmin)
// One 16-row tile per block (A in LDS via async DMA); 8 waves split the 512
// V-tiles in groups of 4: 4 independent WMMA accumulator chains per wave,
// A-fragment reused 4x per LDS read.
// ---------------------------------------------------------------------------
__global__ __launch_bounds__(256) void dist_argmin_kernel(
    const float* __restrict__ X, const float* __restrict__ cb,
    const float* __restrict__ cbsq, int* __restrict__ outIdx)
{
    __shared__ float As[16 * LDA];
    __shared__ float redS[8 * 16];
    __shared__ int   redI[8 * 16];

    const int tid  = threadIdx.x;
    const int wave = tid >> 5;
    const int lane = tid & 31;
    const int m16  = lane & 15;     // A-row / D-column owned by this lane
    const int kh   = lane >> 4;     // K-half (A/B frag) and M-half (C/D)
    const int row0 = blockIdx.x * 16;

    // Stage A tile [16 x 512] -> LDS with async global->LDS b128 DMA
    const uint32_t ldsBase = (uint32_t)(uintptr_t)(&As[0]);
    for (int i = tid; i < 16 * 128; i += 256) {   // 2048 x 16B chunks
        int m  = i >> 7;
        int k4 = (i & 127) << 2;
        async_copy_b128(ldsBase + (uint32_t)((m * LDA + k4) * 4),
                        X + (size_t)(row0 + m) * CC + k4);
    }
    async_wait_all();
    __syncthreads();

    float best[8];
    int   bidx[8];
    #pragma unroll
    for (int j = 0; j < 8; ++j) { best[j] = 3.4e38f; bidx[j] = 0x7FFFFFFF; }

    const float* aptr = &As[m16 * LDA + (kh << 1)];

    for (int vg = wave; vg < VV / 64; vg += 8) {          // uniform per wave
        const int v0 = vg << 6;                           // 4 V-tiles per group
        v8f acc0 = {0.f,0.f,0.f,0.f,0.f,0.f,0.f,0.f};
        v8f acc1 = {0.f,0.f,0.f,0.f,0.f,0.f,0.f,0.f};
        v8f acc2 = {0.f,0.f,0.f,0.f,0.f,0.f,0.f,0.f};
        v8f acc3 = {0.f,0.f,0.f,0.f,0.f,0.f,0.f,0.f};
        const float* b0p = cb + ((size_t)(v0 + m16)) * CC + (kh << 1);
        const float* b1p = b0p + (size_t)16 * CC;
        const float* b2p = b0p + (size_t)32 * CC;
        const float* b3p = b0p + (size_t)48 * CC;
        #pragma unroll 2
        for (int k0 = 0; k0 < CC; k0 += 4) {
            v2f a  = *(const v2f*)(aptr + k0);
            v2f b0 = *(const v2f*)(b0p + k0);
            v2f b1 = *(const v2f*)(b1p + k0);
            v2f b2 = *(const v2f*)(b2p + k0);
            v2f b3 = *(const v2f*)(b3p + k0);
            acc0 = __builtin_amdgcn_wmma_f32_16x16x4_f32(false, a, false, b0, (short)0, acc0, false, false);
            acc1 = __builtin_amdgcn_wmma_f32_16x16x4_f32(false, a, false, b1, (short)0, acc1, false, false);
            acc2 = __builtin_amdgcn_wmma_f32_16x16x4_f32(false, a, false, b2, (short)0, acc2, false, false);
            acc3 = __builtin_amdgcn_wmma_f32_16x16x4_f32(false, a, false, b3, (short)0, acc3, false, false);
        }
        #pragma unroll
        for (int tix = 0; tix < 4; ++tix) {
            const v8f& acc = (tix == 0) ? acc0 : (tix == 1) ? acc1 : (tix == 2) ? acc2 : acc3;
            const int   v   = v0 + (tix << 4) + m16;   // column this lane owns
            const float csq = cbsq[v];
            #pragma unroll
            for (int j = 0; j < 8; ++j) {
                float s = csq - 2.0f * acc[j];
                if (s < best[j] || (s == best[j] && v < bidx[j])) {
                    best[j] = s; bidx[j] = v;
                }
            }
        }
    }

    // Reduce (min, idx) across the 16 lanes of each half (rows j+8*kh)
    #pragma unroll
    for (int off = 8; off >= 1; off >>= 1) {
        #pragma unroll
        for (int j = 0; j < 8; ++j) {
            float os = __shfl_xor(best[j], off, 32);
            int   oi = __shfl_xor(bidx[j], off, 32);
            if (os < best[j] || (os == best[j] && oi < bidx[j])) {
                best[j] = os; bidx[j] = oi;
            }
        }
    }
    if (m16 == 0) {                        // lane 0 -> rows 0..7, lane 16 -> rows 8..15
        #pragma unroll
        for (int j = 0; j < 8; ++j) {
            int r = j + (kh << 3);
            redS[wave * 16 + r] = best[j];
            redI[wave * 16 + r] = bidx[j];
        }
    }
    __syncthreads();
    if (tid < 16) {
        float bs = redS[tid]; int bi = redI[tid];
        for (int w = 1; w < 8; ++w) {
            float s  = redS[w * 16 + tid];
            int   i2 = redI[w * 16 + tid];
            if (s < bs || (s == bs && i2 < bi)) { bs = s; bi = i2; }
        }
        outIdx[row0 + tid] = bi;
    }
}

// ---------------------------------------------------------------------------
// Dequantize + linear interp (align_corners=False): pre[b,c,t]
// ---------------------------------------------------------------------------
__global__ __launch_bounds__(256) void dequant_interp_kernel(
    const int* __restrict__ codeIdx, const float* __restrict__ cb,
    float* __restrict__ pre)
{
    int idx = blockIdx.x * 256 + threadIdx.x;           // over B*C*T
    if (idx >= BB * CC * TT) return;
    int t = idx & (TT - 1);
    int r = idx >> 8;
    int c = r & (CC - 1);
    int b = r >> 9;
    float src = (t + 0.5f) * ((float)QQ / (float)TT) - 0.5f;
    src = fminf(fmaxf(src, 0.0f), (float)(QQ - 1));
    float fi = floorf(src);
    int   i0 = (int)fi;
    int   i1 = min(i0 + 1, QQ - 1);
    float w  = src - fi;
    int c0 = codeIdx[b * QQ + i0];
    int c1 = codeIdx[b * QQ + i1];
    float v0 = cb[(size_t)c0 * CC + c];
    float v1 = cb[(size_t)c1 * CC + c];
    pre[idx] = v0 * (1.0f - w) + v1 * w;
}

// ---------------------------------------------------------------------------
// Conv1d (K=3, pad=1) as 3 WMMA sub-GEMMs (one per tap kk):
//   out[b,o,t] = sum_kk sum_i phi[o,i,kk] * pre[b,i,t+kk-1] + bias[o]
// Block = (b, o-tile). Per tap: phi slice [16 x 512] staged in LDS (shared by
// all 8 waves); each wave computes 2 t-tiles with one shared A-fragment.
// Pad condition is loop-invariant (clamped address x 0/1 mask).
// ---------------------------------------------------------------------------
__global__ __launch_bounds__(256) void conv_wmma_kernel(
    const float* __restrict__ pre, const float* __restrict__ phiW,
    const float* __restrict__ phiB, float* __restrict__ out)
{
    __shared__ float Ph[16 * LDA];

    const int tid  = threadIdx.x;
    const int wave = tid >> 5;
    const int lane = tid & 31;
    const int n16  = lane & 15;     // A-row (M=o) for loads, D-column (N=t) for stores
    const int kh   = lane >> 4;

    const int ot = blockIdx.x & 31;
    const int b  = blockIdx.x >> 5;
    const int o0 = ot << 4;

    const float* preB = pre + (size_t)b * (CC * TT);
    const int tA = (wave)     << 4;   // first t-tile of this wave
    const int tB = (wave + 8) << 4;   // second t-tile of this wave

    v8f acc0 = {0.f,0.f,0.f,0.f,0.f,0.f,0.f,0.f};
    v8f acc1 = {0.f,0.f,0.f,0.f,0.f,0.f,0.f,0.f};

    for (int kk = 0; kk < 3; ++kk) {
        __syncthreads();   // previous tap's compute done before restaging
        // stage phi[:, :, kk] slice for rows o0..o0+15 -> Ph[m][i]
        for (int i = tid; i < 16 * CC; i += 256) {
            int m  = i >> 9;
            int ci = i & (CC - 1);
            Ph[m * LDA + ci] = phiW[(size_t)(o0 + m) * (CC * 3) + ci * 3 + kk];
        }
        __syncthreads();

        const float* ap = &Ph[n16 * LDA + (kh << 1)];
        // loop-invariant pad handling per tile
        int ta = tA + n16 + kk - 1;
        int tb = tB + n16 + kk - 1;
        float mA = (ta >= 0 && ta < TT) ? 1.0f : 0.0f;
        float mB = (tb >= 0 && tb < TT) ? 1.0f : 0.0f;
        int tca = min(max(ta, 0), TT - 1);
        int tcb = min(max(tb, 0), TT - 1);
        const float* bpA = preB + (size_t)(kh << 1) * TT;
        #pragma unroll 2
        for (int k0 = 0; k0 < CC; k0 += 4) {
            v2f a = *(const v2f*)(ap + k0);
            const float* pk = bpA + (size_t)k0 * TT;
            v2f ba, bb;
            ba.x = pk[tca]      * mA;
            ba.y = pk[tca + TT] * mA;
            bb.x = pk[tcb]      * mB;
            bb.y = pk[tcb + TT] * mB;
            acc0 = __builtin_amdgcn_wmma_f32_16x16x4_f32(false, a, false, ba, (short)0, acc0, false, false);
            acc1 = __builtin_amdgcn_wmma_f32_16x16x4_f32(false, a, false, bb, (short)0, acc1, false, false);
        }
    }
    #pragma unroll
    for (int j = 0; j < 8; ++j) {
        int o = o0 + j + (kh << 3);
        float bias = phiB[o];
        float* orow = out + ((size_t)b * CC + o) * TT;
        orow[tA + n16] = acc0[j] + bias;
        orow[tB + n16] = acc1[j] + bias;
    }
}

// ---------------------------------------------------------------------------
// Perplexity: single block, LDS histogram (ds atomics) + log-sum reduce
// ---------------------------------------------------------------------------
__global__ __launch_bounds__(1024) void perplexity_kernel(
    const int* __restrict__ idxOrg, float* __restrict__ outP)
{
    __shared__ int   hist[VV];
    __shared__ float partial[32];
    int tid = threadIdx.x;
    for (int i = tid; i < VV; i += 1024) hist[i] = 0;
    __syncthreads();
    for (int i = tid; i < BB * TT; i += 1024) atomicAdd(&hist[idxOrg[i]], 1);
    __syncthreads();
    float s = 0.0f;
    const float inv = 1.0f / (float)(BB * TT);
    for (int i = tid; i < VV; i += 1024) {
        float p = (float)hist[i] * inv;
        s += p * __logf(p + 1e-7f);
    }
    #pragma unroll
    for (int off = 16; off >= 1; off >>= 1) s += __shfl_xor(s, off, 32);
    if ((tid & 31) == 0) partial[tid >> 5] = s;
    __syncthreads();
    if (tid < 32) {
        float v = partial[tid];
        #pragma unroll
        for (int off = 16; off >= 1; off >>= 1) v += __shfl_xor(v, off, 32);
        if (tid == 0) outP[0] = __expf(-v);
    }
}

// ---------------------------------------------------------------------------
// Host-side orchestration
// ---------------------------------------------------------------------------
extern "C" void kernel_launch(void* const* d_in, const int* in_sizes, int n_in,
                              void* d_out, int out_size, void* d_ws, size_t ws_size,
                              hipStream_t stream) {
    const float* z    = (const float*)d_in[0];   // [B,C,T]
    const float* cb   = (const float*)d_in[1];   // [V,C]
    const float* phiW = (const float*)d_in[2];   // [C,C,3]
    const float* phiB = (const float*)d_in[3];   // [C]
    float* out = (float*)d_out;                  // [B,C,T] z_hat + [1] perplexity

    // Workspace layout (floats)
    float* ws = (float*)d_ws;
    const size_t OFF_X2   = 0;                                    // 8192*512
    const size_t OFF_X1   = OFF_X2 + (size_t)BB * TT * CC;        // 2048*512
    const size_t OFF_PRE  = OFF_X1 + (size_t)BB * QQ * CC;        // B*C*T
    const size_t OFF_CBSQ = OFF_PRE + (size_t)BB * CC * TT;       // V
    const size_t OFF_CIDX = OFF_CBSQ + (size_t)VV;                // B*Q ints
    const size_t OFF_OIDX = OFF_CIDX + (size_t)BB * QQ;           // B*T ints

    float* X2   = ws + OFF_X2;
    float* X1   = ws + OFF_X1;
    float* PRE  = ws + OFF_PRE;
    float* CBSQ = ws + OFF_CBSQ;
    int*   CIDX = (int*)(ws + OFF_CIDX);
    int*   OIDX = (int*)(ws + OFF_OIDX);

    // 1) prep
    downsample_kernel<<<(BB * QQ * CC + 255) / 256, 256, 0, stream>>>(z, X1);
    transpose_kernel <<<(BB * TT * CC + 255) / 256, 256, 0, stream>>>(z, X2);
    cbsq_kernel      <<<VV / 8, 256, 0, stream>>>(cb, CBSQ);

    // 2) argmin over codebook (WMMA f32), q-resolution and full resolution
    dist_argmin_kernel<<<(BB * QQ) / 16, 256, 0, stream>>>(X1, cb, CBSQ, CIDX);
    dist_argmin_kernel<<<(BB * TT) / 16, 256, 0, stream>>>(X2, cb, CBSQ, OIDX);

    // 3) perplexity from full-resolution codes
    perplexity_kernel<<<1, 1024, 0, stream>>>(OIDX, out + (size_t)BB * CC * TT);

    // 4) dequantize + linear interp, then conv1d via WMMA GEMM
    dequant_interp_kernel<<<(BB * CC * TT + 255) / 256, 256, 0, stream>>>(CIDX, cb, PRE);
    conv_wmma_kernel<<<BB * (CC / 16), 256, 0, stream>>>(PRE, phiW, phiB, out);
}